// YoloModel_41145786696156
// MI455X (gfx1250) — compile-verified
//
#include <hip/hip_runtime.h>
#include <stdint.h>

#define N_BOXES   262144
#define N_FEAT    85
#define N_CLASSES 80
#define CONF_THRES 0.8f
#define NMS_THRES  0.4f
#define MAX_DET    300

#define FILTER_ROWS 128                          // boxes per block
#define TILE_ELEMS  (FILTER_ROWS * N_FEAT)       // 10880 dwords = 43520 B

#define NMS_THREADS 1024
#define NWORDS      (N_BOXES / 32)               // 8192 words = 32 KB LDS bitmask

typedef unsigned long long u64;
typedef unsigned int u32;
typedef u32 v4u __attribute__((ext_vector_type(4)));
typedef u32 v8u __attribute__((ext_vector_type(8)));

// monotone float -> uint mapping (ascending order preserved, NaN-free inputs)
__device__ __forceinline__ u32 f32_ord(float f) {
  u32 b = __float_as_uint(f);
  return (b & 0x80000000u) ? ~b : (b | 0x80000000u);
}

__global__ void init_kernel(int* nValid) { *nValid = 0; }

// ---------------------------------------------------------------------------
// Stage 1: filter + score. One TDM tensor_load_to_lds per workgroup DMAs the
// contiguous 128x85-float tile into LDS (TENSORcnt), then each thread reduces
// one row from LDS. D# built per CDNA5 ISA.
// ---------------------------------------------------------------------------
__global__ __launch_bounds__(FILTER_ROWS) void filter_kernel(
    const float* __restrict__ det, u64* __restrict__ key,
    float4* __restrict__ box, float2* __restrict__ confcls,
    int* __restrict__ nValid) {
  __shared__ __align__(16) float tile[TILE_ELEMS];   // 43520 B
  const int tid = threadIdx.x;
  const float* src = det + (long long)blockIdx.x * TILE_ELEMS;
  // low 32 bits of a generic pointer into LDS == byte offset in LDS
  unsigned ldsBase = (unsigned)(uintptr_t)tile;

  if ((tid >> 5) == 0) {                 // wave 0 issues the descriptor
    u64 gaddr = (u64)(uintptr_t)src;
    v4u g0; v8u g1;
    // group0: count=1 | lds_addr | global_addr[56:0] | type=2 ("image")
    g0[0] = 1u;
    g0[1] = ldsBase;
    g0[2] = (u32)gaddr;
    g0[3] = ((u32)(gaddr >> 32) & 0x01FFFFFFu) | (2u << 30);
    // group1: data_size=4B; 1-D tile: tensor_dim0=tile_dim0=10880, dim1=1
    g1[0] = (2u << 16);                               // data_size=2 (4 bytes)
    g1[1] = (u32)(TILE_ELEMS & 0xFFFF) << 16;         // tensor_dim0[15:0]
    g1[2] = (u32)(TILE_ELEMS >> 16) | (1u << 16);     // tensor_dim0[31:16], tensor_dim1=1
    g1[3] = (u32)(TILE_ELEMS & 0xFFFF) << 16;         // tensor_dim1 hi=0, tile_dim0
    g1[4] = 0u;                                       // tile_dim1=0, tile_dim2=0 (1-D)
    g1[5] = (u32)TILE_ELEMS;                          // tensor_dim0_stride[31:0]
    g1[6] = 0u;                                       // stride hi, dim1_stride lo
    g1[7] = 0u;
    asm volatile("tensor_load_to_lds %0, %1\n\t"
                 "s_wait_tensorcnt 0x0"
                 :: "s"(g0), "s"(g1) : "memory");
  }
  __syncthreads();   // all waves observe the DMA'd tile

  // row stride 85 dwords: 85 mod 64 = 21, gcd(21,64)=1 -> conflict-free
  const float* r = tile + tid * N_FEAT;
  float x = r[0], y = r[1], w = r[2], h = r[3], obj = r[4];
  float best = r[5];
  int bi = 0;
  #pragma unroll
  for (int c = 1; c < N_CLASSES; ++c) {
    float s = r[5 + c];
    if (s > best) { best = s; bi = c; }   // strict > == first-occurrence argmax
  }
  int gidx = blockIdx.x * FILTER_ROWS + tid;
  bool valid = (obj >= CONF_THRES);
  float score = obj * best;
  float sortv = valid ? -score : __int_as_float(0x7F800000);  // +inf if invalid
  // ascending sort of (ordKey, idx) == jnp.argsort(where(valid,-score,inf)), stable
  key[gidx] = ((u64)f32_ord(sortv) << 32) | (u32)gidx;
  box[gidx] = make_float4(x - w * 0.5f, y - h * 0.5f, x + w * 0.5f, y + h * 0.5f);
  confcls[gidx] = make_float2(best, (float)bi);
  if (valid) atomicAdd(nValid, 1);
}

// ---------------------------------------------------------------------------
// Stage 2: bitonic sort of 2^18 u64 keys (L2-resident, 2 MB).
// Global passes only for stride >= 1024; strides <= 512 run in LDS.
// Tile in/out uses CDNA5 async LDS<->global DMA (ASYNCcnt).
// ---------------------------------------------------------------------------
__global__ __launch_bounds__(256) void bitonic_global(u64* __restrict__ key,
                                                      int j, int k) {
  int i = blockIdx.x * blockDim.x + threadIdx.x;
  int ixj = i ^ j;
  if (ixj > i) {
    u64 a = key[i], b = key[ixj];
    bool up = ((i & k) == 0);
    if ((a > b) == up) { key[i] = b; key[ixj] = a; }
  }
}

__global__ __launch_bounds__(1024) void bitonic_shared(u64* __restrict__ key,
                                                       int k, int jstart) {
  __shared__ __align__(16) u64 s[1024];
  const int tid = threadIdx.x;
  const int base = blockIdx.x * 1024;
  u64* gsrc = key + base;
  unsigned ldsAddr = (unsigned)(uintptr_t)s + (unsigned)tid * 8u;
  int voff = tid * 8;

  // async global -> LDS, 8 B per lane (8 KB tile), then wait + barrier
  asm volatile("global_load_async_to_lds_b64 %0, %1, %2"
               :: "v"(ldsAddr), "v"(voff), "s"(gsrc) : "memory");
  asm volatile("s_wait_asynccnt 0x0" ::: "memory");
  __syncthreads();

  for (int j = jstart; j > 0; j >>= 1) {
    int ixj = tid ^ j;             // j <= 512: partner stays in tile
    if (ixj > tid) {
      u64 a = s[tid], b = s[ixj];
      bool up = (((base + tid) & k) == 0);
      if ((a > b) == up) { s[tid] = b; s[ixj] = a; }
    }
    __syncthreads();
  }

  // async LDS -> global writeback (loop above ends in a barrier)
  asm volatile("global_store_async_from_lds_b64 %0, %1, %2"
               :: "v"(voff), "v"(ldsAddr), "s"(gsrc) : "memory");
  asm volatile("s_wait_asynccnt 0x0" ::: "memory");
}

// ---------------------------------------------------------------------------
// Stage 3: greedy serial NMS, single workgroup. Validity bitmask (32 KB) and
// all bit ops live in LDS; candidate payloads are L2-resident gathers.
// Then a 512-wide LDS bitonic re-sorts the 300 picks by conf descending
// (tie -> larger slot first == reference's reversed stable ascending sort).
// ---------------------------------------------------------------------------
__global__ __launch_bounds__(NMS_THREADS) void nms_kernel(
    const u64* __restrict__ key, const float4* __restrict__ box,
    const float2* __restrict__ confcls, const int* __restrict__ nValidPtr,
    float* __restrict__ out) {
  __shared__ u32 vmask[NWORDS];
  __shared__ float pconf[MAX_DET];
  __shared__ float pcls[MAX_DET];
  __shared__ int pkv[MAX_DET];
  __shared__ u64 skey[512];
  __shared__ float4 sCurBox;
  __shared__ float sCurCls;
  __shared__ int sFirst;

  const int tid = threadIdx.x;
  const int nv = min(*nValidPtr, N_BOXES);

  // valid bits are exactly sorted ranks [0, nv): all +inf keys sorted last
  for (int wdx = tid; wdx < NWORDS; wdx += NMS_THREADS) {
    int lo = wdx * 32;
    u32 m;
    if (lo + 32 <= nv)      m = 0xFFFFFFFFu;
    else if (lo >= nv)      m = 0u;
    else                    m = (1u << (nv - lo)) - 1u;
    vmask[wdx] = m;
  }
  __syncthreads();

  for (int it = 0; it < MAX_DET; ++it) {
    if (tid == 0) sFirst = 0x7FFFFFFF;
    __syncthreads();
    // lowest nonzero mask word (blocked scan, 8 words/thread)
    const int wpt = NWORDS / NMS_THREADS;
    int local = 0x7FFFFFFF;
    int wbase = tid * wpt;
    for (int q = 0; q < wpt; ++q)
      if (vmask[wbase + q] != 0u) { local = wbase + q; break; }
    if (local != 0x7FFFFFFF) atomicMin(&sFirst, local);
    __syncthreads();

    if (sFirst == 0x7FFFFFFF) {            // uniform: nothing valid left
      for (int r2 = it + tid; r2 < MAX_DET; r2 += NMS_THREADS) {
        pconf[r2] = 0.f; pcls[r2] = 0.f; pkv[r2] = 0;
      }
      __syncthreads();
      break;
    }
    if (tid == 0) {
      int wdx = sFirst;
      int bit = __ffs(vmask[wdx]) - 1;
      int rank = wdx * 32 + bit;           // highest remaining score
      u32 oi = (u32)(key[rank] & 0xFFFFFFFFull);
      float2 cc = confcls[oi];
      sCurBox = box[oi];
      sCurCls = cc.y;
      pconf[it] = cc.x; pcls[it] = cc.y; pkv[it] = 1;
    }
    __syncthreads();

    const float4 cb = sCurBox;
    const float ccls = sCurCls;
    const float a1 = (cb.z - cb.x + 1.f) * (cb.w - cb.y + 1.f);
    for (int r2 = tid; r2 < nv; r2 += NMS_THREADS) {
      u32 m = vmask[r2 >> 5];
      if (!(m & (1u << (r2 & 31)))) continue;   // cheap LDS pre-check
      u32 oi = (u32)(key[r2] & 0xFFFFFFFFull);
      float2 cc = confcls[oi];
      if (cc.y != ccls) continue;               // same float class id only
      float4 b = box[oi];
      float iw = fmaxf(fminf(cb.z, b.z) - fmaxf(cb.x, b.x) + 1.f, 0.f);
      float ih = fmaxf(fminf(cb.w, b.w) - fmaxf(cb.y, b.y) + 1.f, 0.f);
      float inter = iw * ih;
      float a2 = (b.z - b.x + 1.f) * (b.w - b.y + 1.f);
      float iou = inter / (a1 + a2 - inter + 1e-16f);
      if (iou > NMS_THRES)                      // suppress (includes self)
        atomicAnd(&vmask[r2 >> 5], ~(1u << (r2 & 31)));
    }
    __syncthreads();
  }
  __syncthreads();

  // final 512-wide bitonic, descending by (conf-or--inf, slot)
  for (int i2 = tid; i2 < 512; i2 += NMS_THREADS) {
    u64 kk = 0ull;                               // pads sort to the very end
    if (i2 < MAX_DET) {
      float ck = pkv[i2] ? pconf[i2] : __int_as_float(0xFF800000);  // -inf
      kk = ((u64)f32_ord(ck) << 32) | (u32)i2;
    }
    skey[i2] = kk;
  }
  __syncthreads();
  for (int k2 = 2; k2 <= 512; k2 <<= 1) {
    for (int j2 = k2 >> 1; j2 > 0; j2 >>= 1) {
      if (tid < 512) {
        int ixj = tid ^ j2;
        if (ixj > tid) {
          u64 a = skey[tid], b = skey[ixj];
          bool down = ((tid & k2) == 0);
          if ((a < b) == down) { skey[tid] = b; skey[ixj] = a; }
        }
      }
      __syncthreads();
    }
  }
  for (int r2 = tid; r2 < MAX_DET; r2 += NMS_THREADS) {
    int slot = (int)(skey[r2] & 0xFFFFFFFFull);
    out[r2] = pcls[slot];                // ids   (0 for empty picks)
    out[MAX_DET + r2] = pconf[slot];     // probs (0 for empty picks)
  }
}

// ---------------------------------------------------------------------------
extern "C" void kernel_launch(void* const* d_in, const int* in_sizes, int n_in,
                              void* d_out, int out_size, void* d_ws, size_t ws_size,
                              hipStream_t stream) {
  (void)in_sizes; (void)n_in; (void)out_size; (void)ws_size;
  const float* det = (const float*)d_in[0];
  float* out = (float*)d_out;

  // workspace layout: 2 MB keys + 4 MB boxes + 2 MB conf/cls + counter
  u64*    key     = (u64*)d_ws;
  float4* box     = (float4*)(key + N_BOXES);
  float2* confcls = (float2*)(box + N_BOXES);
  int*    nValid  = (int*)(confcls + N_BOXES);

  init_kernel<<<1, 1, 0, stream>>>(nValid);
  filter_kernel<<<N_BOXES / FILTER_ROWS, FILTER_ROWS, 0, stream>>>(
      det, key, box, confcls, nValid);

  for (int k = 2; k <= N_BOXES; k <<= 1) {
    int j = k >> 1;
    while (j >= 1024) {                       // cross-tile strides: global
      bitonic_global<<<N_BOXES / 256, 256, 0, stream>>>(key, j, k);
      j >>= 1;
    }
    bitonic_shared<<<N_BOXES / 1024, 1024, 0, stream>>>(key, k, j);
  }

  nms_kernel<<<1, NMS_THREADS, 0, stream>>>(key, box, confcls, nValid, out);
}